// Cross_Attention_MSA_91139206021773
// MI455X (gfx1250) — compile-verified
//
#include <hip/hip_runtime.h>
#include <hip/hip_bf16.h>
#include <math.h>

typedef __attribute__((ext_vector_type(2))) float v2f;
typedef __attribute__((ext_vector_type(8))) float v8f;
typedef __attribute__((ext_vector_type(4))) int   v4i;

// Problem constants
#define BATCH   4
#define HH      128
#define WW      128
#define NTOK    16384      // per batch
#define TTOK    65536      // total tokens
#define DIM     64
#define HEADS   8
#define DH      64
#define DK      24
#define DHK     3

// chunking for the S = kh^T @ x reduction
#define CHUNK_TOK 128
#define NCHUNK    128      // per batch (128*128 = 16384)

#if defined(__has_builtin)
#  if __has_builtin(__builtin_amdgcn_global_load_async_to_lds_b128)
#    define USE_ASYNC_LDS 1
#  endif
#endif

static __device__ __forceinline__ v8f wmma_f32(v2f a, v2f b, v8f c) {
    return __builtin_amdgcn_wmma_f32_16x16x4_f32(false, a, false, b, (short)0, c, false, false);
}

// 16-byte global -> LDS copy: async on CDNA5 (ASYNCcnt), vector copy otherwise.
static __device__ __forceinline__ void copy16(float* l, const float* g) {
#ifdef USE_ASYNC_LDS
    __builtin_amdgcn_global_load_async_to_lds_b128(
        (__attribute__((address_space(1))) v4i*)(g),
        (__attribute__((address_space(3))) v4i*)(l), 0, 0);
#else
    *(float4*)l = *(const float4*)g;
#endif
}

// wait for async copies (if used) and barrier the block
static __device__ __forceinline__ void tile_join() {
#ifdef USE_ASYNC_LDS
#  if __has_builtin(__builtin_amdgcn_s_wait_asynccnt)
    __builtin_amdgcn_s_wait_asynccnt(0);
#  else
    asm volatile("s_wait_asynccnt 0" ::: "memory");
#  endif
#endif
    __syncthreads();
}

// ---------------------------------------------------------------------------
// Kernel 1: k_inp[t][24] = illu_map[t][24] @ Wk[24][24]   (WMMA, N padded->32)
// grid: TTOK/128 blocks x 256 threads (8 waves, one 16-token tile each)
// ---------------------------------------------------------------------------
__global__ __launch_bounds__(256) void kproj_kernel(
    const float* __restrict__ imap, const float* __restrict__ Wk,
    float* __restrict__ kinp)
{
    __shared__ float sWk[24 * 32];     // [k][n], n padded to 32 with zeros
    __shared__ float sA[128 * 24];     // 128 tokens x 24 ch
    const int tid = threadIdx.x;

    for (int i = tid; i < 24 * 32; i += 256) {
        int r = i >> 5, c = i & 31;
        sWk[i] = (c < 24) ? Wk[r * 24 + c] : 0.0f;
    }
    const int base = blockIdx.x * 128;          // global token base
    // 128*24 floats = 768 x 16B chunks, 3 per thread
    for (int i = tid; i < 768; i += 256)
        copy16(&sA[i * 4], &imap[base * 24 + i * 4]);
    tile_join();

    const int wid = tid >> 5, lane = tid & 31;
    const int ln = lane & 15, half = lane >> 4;
    const int tTok = wid * 16;                  // local token base of this wave

    v8f c0 = {}; v8f c1 = {};
#pragma unroll
    for (int s = 0; s < 6; ++s) {               // K = 24 in steps of 4
        const int kb = s * 4 + half * 2;
        v2f a;  a.x  = sA[(tTok + ln) * 24 + kb];
                a.y  = sA[(tTok + ln) * 24 + kb + 1];
        v2f b0; b0.x = sWk[kb * 32 + ln];        b0.y = sWk[(kb + 1) * 32 + ln];
        v2f b1; b1.x = sWk[kb * 32 + 16 + ln];   b1.y = sWk[(kb + 1) * 32 + 16 + ln];
        c0 = wmma_f32(a, b0, c0);
        c1 = wmma_f32(a, b1, c1);
    }
    const int tok0 = base + tTok;
#pragma unroll
    for (int j = 0; j < 8; ++j) {
        const int row = j + half * 8;           // token within tile
        const int t = tok0 + row;
        kinp[t * 24 + ln] = c0[j];
        const int ch = 16 + ln;
        if (ch < 24) kinp[t * 24 + ch] = c1[j];
    }
}

// ---------------------------------------------------------------------------
// Kernel 2: partial S = kh^T @ x over a 128-token chunk.
// A = kh^T : [24(->32) x 128], B = x : [128 x 64].  grid: BATCH*NCHUNK blocks.
// spart layout: [block][32][64]
// ---------------------------------------------------------------------------
__global__ __launch_bounds__(256) void s_partial_kernel(
    const float* __restrict__ x, const float* __restrict__ kinp,
    float* __restrict__ spart)
{
    __shared__ float sK[CHUNK_TOK * 32];   // [tok][32ch] zero padded
    __shared__ float sX[CHUNK_TOK * 64];   // [tok][64]
    const int tid = threadIdx.x;
    const int b = blockIdx.x >> 7;         // /NCHUNK
    const int chunk = blockIdx.x & 127;
    const int tok0 = b * NTOK + chunk * CHUNK_TOK;

    // data columns 0..23: 128 rows x 6 chunks of 16B (row start = 96B-aligned)
    for (int i = tid; i < 128 * 6; i += 256) {
        const int r = i / 6, c6 = i - r * 6;
        copy16(&sK[r * 32 + c6 * 4], &kinp[(tok0 + r) * 24 + c6 * 4]);
    }
    // zero the pad columns 24..31
    for (int i = tid; i < 128 * 8; i += 256) {
        const int r = i >> 3, c = i & 7;
        sK[r * 32 + 24 + c] = 0.0f;
    }
    // x tile: 128*64 floats = 2048 chunks
    for (int i = tid; i < 2048; i += 256)
        copy16(&sX[i * 4], &x[tok0 * 64 + i * 4]);
    tile_join();

    const int wid = tid >> 5, lane = tid & 31;
    const int Mt = wid >> 2, Nt = wid & 3;       // 2 x 4 tiles = 8 waves
    const int ln = lane & 15, half = lane >> 4;

    v8f acc = {};
#pragma unroll 8
    for (int s = 0; s < CHUNK_TOK / 4; ++s) {    // K = 128 tokens, steps of 4
        const int kb = s * 4 + half * 2;
        v2f a;  a.x = sK[kb * 32 + Mt * 16 + ln];
                a.y = sK[(kb + 1) * 32 + Mt * 16 + ln];
        v2f bb; bb.x = sX[kb * 64 + Nt * 16 + ln];
                bb.y = sX[(kb + 1) * 64 + Nt * 16 + ln];
        acc = wmma_f32(a, bb, acc);
    }
    float* dst = spart + blockIdx.x * (32 * 64);
#pragma unroll
    for (int j = 0; j < 8; ++j) {
        const int row = Mt * 16 + j + half * 8;
        const int col = Nt * 16 + ln;
        dst[row * 64 + col] = acc[j];
    }
}

// ---------------------------------------------------------------------------
// Kernel 3: deterministic reduction of S partials over NCHUNK chunks.
// S[b][24][64].  grid: BATCH*24*64/256 = 24 blocks.
// ---------------------------------------------------------------------------
__global__ __launch_bounds__(256) void s_reduce_kernel(
    const float* __restrict__ spart, float* __restrict__ S)
{
    const int e = blockIdx.x * 256 + threadIdx.x;   // < 4*1536
    const int b = e / 1536;
    const int rc = e - b * 1536;                    // row*64+col, row<24
    float sum = 0.0f;
    for (int ch = 0; ch < NCHUNK; ++ch)
        sum += spart[(b * NCHUNK + ch) * (32 * 64) + rc];
    S[e] = sum;
}

// ---------------------------------------------------------------------------
// Kernel 4: per-batch tiny kernel:
//   attn = (S @ Wq_h) * (1/8) * rescale[h]; softmax over d;
//   M[c][kk*8+h] = sum_d attn[h*3+kk][d] * Wv[c][h*64+d];
//   P[c][j] = sum_t M[c][t] * Wp[t][j]   (P padded to 32 cols)
// grid: BATCH blocks x 64 threads.
// ---------------------------------------------------------------------------
__global__ __launch_bounds__(64) void attn_kernel(
    const float* __restrict__ S, const float* __restrict__ Wq,
    const float* __restrict__ Wv, const float* __restrict__ Wp,
    const float* __restrict__ rescale, float* __restrict__ P)
{
    __shared__ float sS[24 * 64];
    __shared__ float sAttn[24 * 64];
    const int b = blockIdx.x;
    const int tid = threadIdx.x;

    for (int i = tid; i < 24 * 64; i += 64) sS[i] = S[b * 1536 + i];
    __syncthreads();

    if (tid < 24) {
        const int r = tid;                 // r = h*3 + kk
        const int h = r / 3;
        const float scale = 0.125f * rescale[h];
        for (int d = 0; d < 64; ++d) {
            float v = 0.0f;
            for (int c = 0; c < 64; ++c)
                v += sS[r * 64 + c] * Wq[c * 512 + h * 64 + d];
            sAttn[r * 64 + d] = v * scale;
        }
        float mx = -1e30f;
        for (int d = 0; d < 64; ++d) mx = fmaxf(mx, sAttn[r * 64 + d]);
        float sm = 0.0f;
        for (int d = 0; d < 64; ++d) {
            float ev = __expf(sAttn[r * 64 + d] - mx);
            sAttn[r * 64 + d] = ev;
            sm += ev;
        }
        const float inv = 1.0f / sm;
        for (int d = 0; d < 64; ++d) sAttn[r * 64 + d] *= inv;
    }
    __syncthreads();

    const int c = tid;                     // one channel row of P per thread
    float Mv[24];
#pragma unroll
    for (int t = 0; t < 24; ++t) {         // t = kk*8 + h  (output flat order)
        const int kk = t >> 3, h = t & 7;
        float v = 0.0f;
        for (int d = 0; d < 64; ++d)
            v += sAttn[(h * 3 + kk) * 64 + d] * Wv[c * 512 + h * 64 + d];
        Mv[t] = v;
    }
    for (int j = 0; j < 32; ++j) {
        float v = 0.0f;
        if (j < 24) {
#pragma unroll
            for (int t = 0; t < 24; ++t) v += Mv[t] * Wp[t * 24 + j];
        }
        P[(b * 64 + c) * 32 + j] = v;
    }
}

// ---------------------------------------------------------------------------
// Kernel 5: out[t][j] = fea[t][:] @ P[b] + bp[j]   (WMMA)
// grid: TTOK/128 blocks x 256 threads.
// ---------------------------------------------------------------------------
__global__ __launch_bounds__(256) void out_kernel(
    const float* __restrict__ fea, const float* __restrict__ P,
    const float* __restrict__ bp, float* __restrict__ out)
{
    __shared__ float sF[128 * 64];         // 32 KB
    __shared__ float sP[64 * 32];          // 8 KB
    const int tid = threadIdx.x;
    const int base = blockIdx.x * 128;
    const int b = base >> 14;              // /16384 (blocks never straddle batch)

    for (int i = tid; i < 512; i += 256)
        copy16(&sP[i * 4], &P[b * 2048 + i * 4]);
    for (int i = tid; i < 2048; i += 256)
        copy16(&sF[i * 4], &fea[base * 64 + i * 4]);
    tile_join();

    const int wid = tid >> 5, lane = tid & 31;
    const int ln = lane & 15, half = lane >> 4;
    const int tTok = wid * 16;

    v8f c0 = {}; v8f c1 = {};
#pragma unroll
    for (int s = 0; s < 16; ++s) {         // K = 64
        const int kb = s * 4 + half * 2;
        v2f a;  a.x  = sF[(tTok + ln) * 64 + kb];
                a.y  = sF[(tTok + ln) * 64 + kb + 1];
        v2f b0; b0.x = sP[kb * 32 + ln];       b0.y = sP[(kb + 1) * 32 + ln];
        v2f b1; b1.x = sP[kb * 32 + 16 + ln];  b1.y = sP[(kb + 1) * 32 + 16 + ln];
        c0 = wmma_f32(a, b0, c0);
        c1 = wmma_f32(a, b1, c1);
    }
    const int tok0 = base + tTok;
#pragma unroll
    for (int j = 0; j < 8; ++j) {
        const int row = j + half * 8;
        const int t = tok0 + row;
        out[t * 24 + ln] = c0[j] + bp[ln];
        const int ch = 16 + ln;
        if (ch < 24) out[t * 24 + ch] = c1[j] + bp[ch];
    }
}

// ---------------------------------------------------------------------------
// Kernels 6/7: grouped 3x3 conv (groups=3, 8 ch/group), NHWC in/out.
// conv1: + exact GELU, writes y1. conv2: accumulates into out.
// grid: TTOK/256 blocks x 256 threads; one token per thread.
// ---------------------------------------------------------------------------
template <bool GELU, bool ACCUM>
__global__ __launch_bounds__(256) void gconv_kernel(
    const float* __restrict__ in, const float* __restrict__ w,
    float* __restrict__ out)
{
    __shared__ float sw[24 * 8 * 9];       // [o][ic][dy][dx], matches OIHW flat
    const int tid = threadIdx.x;
    for (int i = tid; i < 24 * 8 * 9; i += 256) sw[i] = w[i];
    __syncthreads();

    const int t = blockIdx.x * 256 + tid;  // global token
    const int pos = t & (NTOK - 1);
    const int y = pos >> 7;
    const int x = pos & 127;

    float acc[24];
#pragma unroll
    for (int o = 0; o < 24; ++o) acc[o] = 0.0f;

    for (int dy = 0; dy < 3; ++dy) {
        const int yy = y + dy - 1;
        if (yy < 0 || yy > 127) continue;
        for (int dx = 0; dx < 3; ++dx) {
            const int xx = x + dx - 1;
            if (xx < 0 || xx > 127) continue;
            const float* p = in + (t + (dy - 1) * 128 + (dx - 1)) * 24;
            float v[24];
#pragma unroll
            for (int c = 0; c < 24; ++c) v[c] = p[c];
#pragma unroll
            for (int o = 0; o < 24; ++o) {
                const int gb = (o >> 3) * 8;
                float s = 0.0f;
#pragma unroll
                for (int ic = 0; ic < 8; ++ic)
                    s += v[gb + ic] * sw[o * 72 + ic * 9 + dy * 3 + dx];
                acc[o] += s;
            }
        }
    }
#pragma unroll
    for (int o = 0; o < 24; ++o) {
        float r = acc[o];
        if (GELU) r = 0.5f * r * (1.0f + erff(r * 0.70710678118654752f));
        if (ACCUM) out[t * 24 + o] = out[t * 24 + o] + r;
        else       out[t * 24 + o] = r;
    }
}

// ---------------------------------------------------------------------------
extern "C" void kernel_launch(void* const* d_in, const int* in_sizes, int n_in,
                              void* d_out, int out_size, void* d_ws, size_t ws_size,
                              hipStream_t stream) {
    const float* x_in    = (const float*)d_in[0];   // [4,128,128,64]
    const float* fea     = (const float*)d_in[1];   // [4,128,128,64]
    const float* imap    = (const float*)d_in[2];   // [4,128,128,24]
    const float* Wq      = (const float*)d_in[3];   // [64,512]
    const float* Wk      = (const float*)d_in[4];   // [24,24]
    const float* Wv      = (const float*)d_in[5];   // [64,512]
    const float* rescale = (const float*)d_in[6];   // [8]
    const float* Wp      = (const float*)d_in[7];   // [24,24]
    const float* bp      = (const float*)d_in[8];   // [24]
    const float* c1w     = (const float*)d_in[9];   // [24,8,3,3]
    const float* c2w     = (const float*)d_in[10];  // [24,8,3,3]
    float* out = (float*)d_out;                     // [4,128,128,24]

    float* ws    = (float*)d_ws;
    float* kinp  = ws;                               // TTOK*24        = 1572864
    float* y1    = kinp  + (size_t)TTOK * 24;        // TTOK*24        = 1572864
    float* spart = y1    + (size_t)TTOK * 24;        // B*NCHUNK*32*64 = 1048576
    float* S     = spart + (size_t)BATCH * NCHUNK * 32 * 64;  // 6144
    float* P     = S     + (size_t)BATCH * 24 * 64;  // B*64*32 = 8192

    kproj_kernel    <<<TTOK / 128, 256, 0, stream>>>(imap, Wk, kinp);
    s_partial_kernel<<<BATCH * NCHUNK, 256, 0, stream>>>(x_in, kinp, spart);
    s_reduce_kernel <<<(BATCH * 24 * 64) / 256, 256, 0, stream>>>(spart, S);
    attn_kernel     <<<BATCH, 64, 0, stream>>>(S, Wq, Wv, Wp, rescale, P);
    out_kernel      <<<TTOK / 128, 256, 0, stream>>>(fea, P, bp, out);
    gconv_kernel<true,  false><<<TTOK / 256, 256, 0, stream>>>(kinp, c1w, y1);
    gconv_kernel<false, true> <<<TTOK / 256, 256, 0, stream>>>(y1, c2w, out);
}